// Generator_24197845745924
// MI455X (gfx1250) — compile-verified
//
#include <hip/hip_runtime.h>
#include <hip/hip_bf16.h>

// Problem constants (from reference)
#define Bb 32
#define Ee 256
#define Rr 256
#define Hh 512
#define Vv 50257
#define Tt 128
#define KX 512   // E+R == H == 512: K dim for both x and h passes

typedef __attribute__((ext_vector_type(2))) float v2f;
typedef __attribute__((ext_vector_type(8))) float v8f;

__device__ __forceinline__ float sigf(float x) { return 1.0f / (1.0f + __expf(-x)); }

// ---------------------------------------------------------------------------
// LSTM cell: one wave (32 threads) per block; block computes a 16x16 tile of
// the hidden state for all four gates using V_WMMA_F32_16X16X4_F32, then the
// fused elementwise state update. grid = 64 blocks (2 m-tiles x 32 n-tiles).
// h is double-buffered (h_in -> h_out); c updated in place (unique ownership).
// ---------------------------------------------------------------------------
__global__ void lstm_cell_kernel(const float* __restrict__ x,
                                 const float* __restrict__ h_in,
                                 float* __restrict__ h_out,
                                 float* __restrict__ c,
                                 const float* __restrict__ Wih,
                                 const float* __restrict__ Whh,
                                 const float* __restrict__ bih,
                                 const float* __restrict__ bhh)
{
    const int lane  = threadIdx.x;        // 0..31 (wave32)
    const int ntile = blockIdx.x & 31;    // 32 n-tiles over H=512
    const int mt    = blockIdx.x >> 5;    // 2 m-tiles over B=32
    const int n0 = ntile * 16;
    const int m0 = mt * 16;
    const int nl = lane & 15;
    const int hi = lane >> 4;
    const int kb = hi * 2;                // K sub-offset per ISA A/B layout

    v8f acc0 = {}, acc1 = {}, acc2 = {}, acc3 = {};

    // pass 0: x @ Wih^T ; pass 1: h @ Whh^T   (both K = 512)
    for (int pass = 0; pass < 2; ++pass) {
        const float* X = pass ? h_in : x;
        const float* W = pass ? Whh : Wih;
        #pragma unroll 4
        for (int k0 = 0; k0 < KX; k0 += 4) {
            // A tile 16x4: lane holds A[m][kb], A[m][kb+1]
            v2f a  = *(const v2f*)(X + (size_t)(m0 + nl) * KX + k0 + kb);
            // B tiles 4x16 (B[k][n] = W[n][k]): one per gate
            v2f b0 = *(const v2f*)(W + (size_t)(0 * Hh + n0 + nl) * KX + k0 + kb);
            v2f b1 = *(const v2f*)(W + (size_t)(1 * Hh + n0 + nl) * KX + k0 + kb);
            v2f b2 = *(const v2f*)(W + (size_t)(2 * Hh + n0 + nl) * KX + k0 + kb);
            v2f b3 = *(const v2f*)(W + (size_t)(3 * Hh + n0 + nl) * KX + k0 + kb);
            acc0 = __builtin_amdgcn_wmma_f32_16x16x4_f32(false, a, false, b0, (short)0, acc0, false, false);
            acc1 = __builtin_amdgcn_wmma_f32_16x16x4_f32(false, a, false, b1, (short)0, acc1, false, false);
            acc2 = __builtin_amdgcn_wmma_f32_16x16x4_f32(false, a, false, b2, (short)0, acc2, false, false);
            acc3 = __builtin_amdgcn_wmma_f32_16x16x4_f32(false, a, false, b3, (short)0, acc3, false, false);
        }
    }

    const int n = n0 + nl;
    const float bI = bih[0 * Hh + n] + bhh[0 * Hh + n];
    const float bF = bih[1 * Hh + n] + bhh[1 * Hh + n];
    const float bG = bih[2 * Hh + n] + bhh[2 * Hh + n];
    const float bO = bih[3 * Hh + n] + bhh[3 * Hh + n];

    #pragma unroll
    for (int r = 0; r < 8; ++r) {
        const int m = m0 + hi * 8 + r;           // batch row (C/D layout)
        const size_t idx = (size_t)m * Hh + n;
        const float ig = sigf(acc0[r] + bI);
        const float fg = sigf(acc1[r] + bF);
        const float gg = tanhf(acc2[r] + bG);
        const float og = sigf(acc3[r] + bO);
        const float cn = fg * c[idx] + ig * gg;
        c[idx] = cn;
        h_out[idx] = og * tanhf(cn);
    }
}

// ---------------------------------------------------------------------------
// FC head: logits = h1 @ Wfc^T + bfc, written to d_out[b][t][:].
// One wave per 16-column tile of V; two accumulators cover all 32 rows,
// reusing the B tile (Wfc stays L2-resident: 103 MB < 192 MB L2).
// ---------------------------------------------------------------------------
__global__ void fc_kernel(const float* __restrict__ h1,
                          const float* __restrict__ Wfc,
                          const float* __restrict__ bfc,
                          float* __restrict__ out_t)   // d_out + t*V, batch stride T*V
{
    const int lane = threadIdx.x;
    const int n0 = blockIdx.x * 16;
    const int nl = lane & 15;
    const int hi = lane >> 4;
    const int kb = hi * 2;
    const int n  = n0 + nl;
    const int nc = (n < Vv) ? n : (Vv - 1);      // clamp tail tile (not stored)

    const float* Wrow = Wfc + (size_t)nc * Hh;
    v8f accA = {}, accB = {};

    #pragma unroll 4
    for (int k0 = 0; k0 < Hh; k0 += 4) {
        v2f a0 = *(const v2f*)(h1 + (size_t)nl * Hh + k0 + kb);          // rows 0..15
        v2f a1 = *(const v2f*)(h1 + (size_t)(nl + 16) * Hh + k0 + kb);   // rows 16..31
        v2f b  = *(const v2f*)(Wrow + k0 + kb);
        accA = __builtin_amdgcn_wmma_f32_16x16x4_f32(false, a0, false, b, (short)0, accA, false, false);
        accB = __builtin_amdgcn_wmma_f32_16x16x4_f32(false, a1, false, b, (short)0, accB, false, false);
    }

    if (n < Vv) {
        const float bias = bfc[n];
        #pragma unroll
        for (int r = 0; r < 8; ++r) {
            const int mA = hi * 8 + r;           // rows 0..15
            const int mB = 16 + hi * 8 + r;      // rows 16..31
            out_t[(size_t)mA * (size_t)Tt * Vv + n] = accA[r] + bias;
            out_t[(size_t)mB * (size_t)Tt * Vv + n] = accB[r] + bias;
        }
    }
}

// ---------------------------------------------------------------------------
// Per-batch-row argmax over V (first-occurrence tie-break, matching
// jnp.argmax), then build next z = concat(emb_table[tok], noise[t]).
// ---------------------------------------------------------------------------
__global__ void argmax_embed_kernel(const float* __restrict__ out_t,   // d_out + t*V
                                    const float* __restrict__ emb,     // [V, E]
                                    const float* __restrict__ noise_t, // [B, R]
                                    float* __restrict__ z)             // [B, E+R]
{
    const int b = blockIdx.x;
    const int tid = threadIdx.x;                 // 256 threads
    const float* row = out_t + (size_t)b * (size_t)Tt * Vv;

    float best = -__builtin_inff();
    int bidx = 0;
    for (int i = tid; i < Vv; i += 256) {
        const float v = row[i];
        if (v > best) { best = v; bidx = i; }    // strictly > keeps first occurrence
    }

    __shared__ float sval[256];
    __shared__ int   sidx[256];
    sval[tid] = best; sidx[tid] = bidx;
    __syncthreads();
    for (int s = 128; s > 0; s >>= 1) {
        if (tid < s) {
            const float v2 = sval[tid + s];
            const int   i2 = sidx[tid + s];
            if (v2 > sval[tid] || (v2 == sval[tid] && i2 < sidx[tid])) {
                sval[tid] = v2; sidx[tid] = i2;
            }
        }
        __syncthreads();
    }
    const int tok = sidx[0];

    // tid == 0..255 == E == R
    z[(size_t)b * (Ee + Rr) + tid]      = emb[(size_t)tok * Ee + tid];
    z[(size_t)b * (Ee + Rr) + Ee + tid] = noise_t[(size_t)b * Rr + tid];
}

// ---------------------------------------------------------------------------
// z0 = concat(keyword, noise0); zero the recurrent states.
// ---------------------------------------------------------------------------
__global__ void init_kernel(const float* __restrict__ keyword,
                            const float* __restrict__ noise0,
                            float* __restrict__ z,
                            float* __restrict__ h0, float* __restrict__ c0,
                            float* __restrict__ h1, float* __restrict__ c1)
{
    const int i = blockIdx.x * blockDim.x + threadIdx.x;
    if (i < Bb * (Ee + Rr)) {
        const int b = i / (Ee + Rr), j = i % (Ee + Rr);
        z[i] = (j < Ee) ? keyword[b * Ee + j] : noise0[b * Rr + (j - Ee)];
        h0[i] = 0.f; c0[i] = 0.f; h1[i] = 0.f; c1[i] = 0.f;
    }
}

extern "C" void kernel_launch(void* const* d_in, const int* in_sizes, int n_in,
                              void* d_out, int out_size, void* d_ws, size_t ws_size,
                              hipStream_t stream)
{
    const float* keyword = (const float*)d_in[0];
    const float* noise0  = (const float*)d_in[1];
    const float* noise   = (const float*)d_in[2];   // [T, B, R]
    const float* emb     = (const float*)d_in[3];   // [V, E]
    const float* W_ih0   = (const float*)d_in[4];
    const float* W_hh0   = (const float*)d_in[5];
    const float* b_ih0   = (const float*)d_in[6];
    const float* b_hh0   = (const float*)d_in[7];
    const float* W_ih1   = (const float*)d_in[8];
    const float* W_hh1   = (const float*)d_in[9];
    const float* b_ih1   = (const float*)d_in[10];
    const float* b_hh1   = (const float*)d_in[11];
    const float* W_fc    = (const float*)d_in[12];
    const float* b_fc    = (const float*)d_in[13];
    float* out = (float*)d_out;                     // [B, T, V]

    float* ws = (float*)d_ws;
    const size_t S = (size_t)Bb * Hh;               // 16384 floats per state buffer
    float* z   = ws;
    float* h0a = ws + 1 * S;
    float* h0b = ws + 2 * S;
    float* c0  = ws + 3 * S;
    float* h1a = ws + 4 * S;
    float* h1b = ws + 5 * S;
    float* c1  = ws + 6 * S;

    init_kernel<<<(Bb * Hh + 255) / 256, 256, 0, stream>>>(keyword, noise0, z, h0a, c0, h1a, c1);

    const int NT_FC = (Vv + 15) / 16;               // 3142 column tiles
    for (int t = 0; t < Tt; ++t) {
        const float* h0_in = (t & 1) ? h0b : h0a;
        float*       h0_o  = (t & 1) ? h0a : h0b;
        const float* h1_in = (t & 1) ? h1b : h1a;
        float*       h1_o  = (t & 1) ? h1a : h1b;

        lstm_cell_kernel<<<64, 32, 0, stream>>>(z, h0_in, h0_o, c0, W_ih0, W_hh0, b_ih0, b_hh0);
        lstm_cell_kernel<<<64, 32, 0, stream>>>(h0_o, h1_in, h1_o, c1, W_ih1, W_hh1, b_ih1, b_hh1);
        fc_kernel<<<NT_FC, 32, 0, stream>>>(h1_o, W_fc, b_fc, out + (size_t)t * Vv);
        argmax_embed_kernel<<<Bb, 256, 0, stream>>>(out + (size_t)t * Vv, emb,
                                                    noise + (size_t)t * Bb * Rr, z);
    }
}